// SoftDTW_63307817943573
// MI455X (gfx1250) — compile-verified
//
#include <hip/hip_runtime.h>
#include <math.h>

// ---------------------------------------------------------------------------
// Soft-DTW forward, MI455X (gfx1250, wave32).
//   Phase 1: D[b,i,j] = sum_h |X[b,i,h]-Y[b,j,h]|  (L1), tiled via async
//            global->LDS copies + register-blocked VALU. D lives in L2 (8MB).
//   Phase 2: anti-diagonal soft-min DP, 1 workgroup per batch, LDS rolling
//            diagonals, one barrier per diagonal step.
// ---------------------------------------------------------------------------

#define B_  8
#define N_  512
#define M_  512
#define H_  80
#define TILE 64
#define LSTR 81            // LDS row stride (odd -> conflict-free b32 reads)
#define DIAG (N_ + 2)      // 514 DP lanes per diagonal

// ---- CDNA5 async global->LDS path (guarded; fallback = plain copy) --------
#if defined(__AMDGCN__) && __has_builtin(__builtin_amdgcn_global_load_async_to_lds_b32)
#define USE_ASYNC_LDS 1
#else
#define USE_ASYNC_LDS 0
#endif

__device__ __forceinline__ void g2lds_b32(const float* gsrc, float* ldst) {
#if USE_ASYNC_LDS
  __builtin_amdgcn_global_load_async_to_lds_b32(
      (__attribute__((address_space(1))) int*)gsrc,
      (__attribute__((address_space(3))) int*)ldst,
      /*offset=*/0, /*cpol=*/0);
#else
  *ldst = *gsrc;
#endif
}

__device__ __forceinline__ void wait_async_lds() {
#if USE_ASYNC_LDS
#if __has_builtin(__builtin_amdgcn_s_wait_asynccnt)
  __builtin_amdgcn_s_wait_asynccnt(0);
#else
  asm volatile("s_wait_asynccnt 0" ::: "memory");
#endif
#endif
}

// ---------------------------------------------------------------------------
// Phase 1: pairwise L1 distance tiles.
// grid = (M/TILE, N/TILE, B) = (8,8,8); block = 256 (8 wave32).
// LDS: 2 * 64*81*4B = 41.5 KB  (WGP has 320 KB).
// Each thread: 4x4 register tile; per h: 8 ds_load_b32 + 32 VALU.
// ---------------------------------------------------------------------------
__global__ __launch_bounds__(256) void sdtw_dist_kernel(
    const float* __restrict__ X, const float* __restrict__ Y,
    float* __restrict__ D) {
  __shared__ float Xs[TILE * LSTR];
  __shared__ float Ys[TILE * LSTR];

  const int b  = blockIdx.z;
  const int i0 = blockIdx.y * TILE;
  const int j0 = blockIdx.x * TILE;
  const float* Xb = X + ((size_t)b * N_ + i0) * H_;
  const float* Yb = Y + ((size_t)b * M_ + j0) * H_;

  // Stage tiles: 4 threads per row, 20 contiguous floats each, async to LDS.
  const int t  = threadIdx.x;
  const int r  = t >> 2;          // row 0..63
  const int hc = (t & 3) * 20;    // 0,20,40,60
#pragma unroll
  for (int e = 0; e < 20; ++e) {
    g2lds_b32(&Xb[r * H_ + hc + e], &Xs[r * LSTR + hc + e]);
    g2lds_b32(&Yb[r * H_ + hc + e], &Ys[r * LSTR + hc + e]);
  }
  wait_async_lds();
  __syncthreads();

  const int tx = t & 15;   // 16 cols of threads
  const int ty = t >> 4;   // 16 rows of threads
  float acc[4][4];
#pragma unroll
  for (int a = 0; a < 4; ++a)
#pragma unroll
    for (int c = 0; c < 4; ++c) acc[a][c] = 0.0f;

#pragma unroll 4
  for (int h = 0; h < H_; ++h) {
    float xv[4], yv[4];
#pragma unroll
    for (int a = 0; a < 4; ++a) xv[a] = Xs[(ty * 4 + a) * LSTR + h];
#pragma unroll
    for (int c = 0; c < 4; ++c) yv[c] = Ys[(tx * 4 + c) * LSTR + h];
#pragma unroll
    for (int a = 0; a < 4; ++a)
#pragma unroll
      for (int c = 0; c < 4; ++c) acc[a][c] += fabsf(xv[a] - yv[c]);
  }

#pragma unroll
  for (int a = 0; a < 4; ++a) {
    float* drow = D + ((size_t)(b * N_ + i0 + ty * 4 + a)) * M_ + j0 + tx * 4;
    float4 v = make_float4(acc[a][0], acc[a][1], acc[a][2], acc[a][3]);
    *reinterpret_cast<float4*>(drow) = v;  // 16B aligned global_store_b128
  }
}

// ---------------------------------------------------------------------------
// Phase 2: soft-min DP over anti-diagonals (gamma = 1, warp = 1).
// Recurrence (cost charged at *source* cell, per reference):
//   R[i,j] = softmin( R[i-1,j-1] + Dp[i-1,j-1],
//                     R[i-1,j]   + Dp[i-1,j]   + warp,
//                     R[i,j-1]   + Dp[i,j-1]   + warp )
// on the (N+2)x(M+2) zero-padded grid; output = R[N+1, M+1].
// grid = B; block = 544 (17 wave32); thread i owns row i (i < 514).
// LDS: 3 R diagonals + 3 Dp diagonals, rotated; 1 barrier per step.
// ---------------------------------------------------------------------------
__global__ __launch_bounds__(544) void sdtw_dp_kernel(
    const float* __restrict__ D, float* __restrict__ out) {
  __shared__ float Rb[3][DIAG + 2];
  __shared__ float Dd[3][DIAG + 2];

  const int b = blockIdx.x;
  const int i = threadIdx.x;
  const bool rowOK = (i < DIAG);
  const float INF = __builtin_inff();

  if (rowOK) {
    Rb[0][i] = (i == 0) ? 0.0f : INF;  // diag k=0
    Rb[1][i] = INF;                    // diag k=1
    Dd[0][i] = 0.0f;                   // Dp on k=0 is all zeros
    Dd[1][i] = 0.0f;                   // Dp on k=1 is all zeros
  }
  const float* Dg = D + (size_t)b * N_ * M_;

  auto loadD = [&](int ii, int jj) -> float {
    if (ii >= 1 && ii <= N_ && jj >= 1 && jj <= M_)
      return Dg[(size_t)(ii - 1) * M_ + (jj - 1)];
    return 0.0f;  // zero-padded border rows/cols
  };

  // Software pipeline: d_next holds Dp[i, k-i] for the upcoming step.
  float d_next = rowOK ? loadD(i, 2 - i) : 0.0f;
  float d_own_prev = 0.0f;  // Dp[i, (k-1)-i]; diag k=1 is all zeros
  __syncthreads();

  const int kLast = 2 * N_ + 2;  // 1026
  for (int k = 2; k <= kLast; ++k) {
    const int j = k - i;
    const float d_cur = d_next;
    d_next = rowOK ? loadD(i, j + 1) : 0.0f;  // prefetch for step k+1 (L2 hit)

    float rval = INF;
    const bool valid = rowOK && (i >= 1) && (j >= 1) && (j <= M_ + 1);
    if (valid) {
      const float rm2  = Rb[(k + 1) % 3][i - 1];  // R[i-1, j-1]
      const float rm1a = Rb[(k + 2) % 3][i - 1];  // R[i-1, j]
      const float rm1b = Rb[(k + 2) % 3][i];      // R[i,   j-1]
      const float dm2  = Dd[(k + 1) % 3][i - 1];  // Dp[i-1, j-1]
      const float dm1a = Dd[(k + 2) % 3][i - 1];  // Dp[i-1, j]
      // gamma == 1, warp == 1
      const float r0 = -(rm2 + dm2);
      const float r1 = -(rm1a + dm1a + 1.0f);
      const float r2 = -(rm1b + d_own_prev + 1.0f);
      const float m = fmaxf(r0, fmaxf(r1, r2));
      if (m > -INF) {  // all three -inf -> stays +INF (boundary semantics)
        const float s = __expf(r0 - m) + __expf(r1 - m) + __expf(r2 - m);
        rval = -(m + __logf(s));
      }
    }
    if (rowOK) {
      Rb[k % 3][i] = rval;   // overwrite diag k-3 slot
      Dd[k % 3][i] = d_cur;  // publish Dp[i, k-i] for neighbors next steps
    }
    d_own_prev = d_cur;
    __syncthreads();
  }

  if (i == N_ + 1) out[b] = Rb[kLast % 3][N_ + 1];  // R[N+1, M+1]
}

// ---------------------------------------------------------------------------
extern "C" void kernel_launch(void* const* d_in, const int* in_sizes, int n_in,
                              void* d_out, int out_size, void* d_ws, size_t ws_size,
                              hipStream_t stream) {
  (void)in_sizes; (void)n_in; (void)out_size; (void)ws_size;
  const float* X = (const float*)d_in[0];  // (8, 512, 80) f32
  const float* Y = (const float*)d_in[1];  // (8, 512, 80) f32
  float* D   = (float*)d_ws;               // 8*512*512*4 = 8 MB scratch
  float* out = (float*)d_out;              // (8,) f32

  dim3 g1(M_ / TILE, N_ / TILE, B_);       // 8 x 8 x 8 = 512 workgroups
  sdtw_dist_kernel<<<g1, 256, 0, stream>>>(X, Y, D);
  sdtw_dp_kernel<<<B_, 544, 0, stream>>>(D, out);
}